// WeightedCrossEntropyLossLearnable_5317169513158
// MI455X (gfx1250) — compile-verified
//
#include <hip/hip_runtime.h>
#include <hip/hip_bf16.h>
#include <stdint.h>

// ---------------- problem constants ----------------
#define N_SAMPLES 4194304
#define NCLS 5
#define MIN_VALUE 0.5f

#define THREADS 256
#define NBLOCKS 1024
#define SAMPLES_PER_BLOCK (N_SAMPLES / NBLOCKS)                  // 4096
#define SAMPLES_PER_TILE  1024
#define TILE_ELEMS        (SAMPLES_PER_TILE * NCLS)              // 5120 floats = 20 KB
#define TILES_PER_BLOCK   (SAMPLES_PER_BLOCK / SAMPLES_PER_TILE) // 4
#define SAMPLES_PER_THREAD (SAMPLES_PER_TILE / THREADS)          // 4

typedef unsigned int v4u __attribute__((ext_vector_type(4)));
typedef int          v8i __attribute__((ext_vector_type(8)));
typedef int          v4i __attribute__((ext_vector_type(4)));
typedef float        v8f __attribute__((ext_vector_type(8)));
typedef float        v2f __attribute__((ext_vector_type(2)));

struct Smem {
    float pred[2][TILE_ELEMS];   // double-buffered y_pred tiles (TDM destination)
    float ratio[32];             // clamped 5x5 adjustment matrix
    float wred[16];              // per-wave partials (8 numer + 8 denom)
};

// Issue one 1-D TDM tile load: TILE_ELEMS f32 from gsrc -> LDS byte offset.
// D# encoding per CDNA5 ISA ch.8 (group0: count/lds_addr/global_addr/type=2;
// group1: data_size=4B, tensor_dim0=tile_dim0=TILE_ELEMS, 1-D tile).
// amdgpu-toolchain (clang-23) builtin: 6 args, trailing int32x8 zero-filled.
__device__ __forceinline__ void tdm_load_tile(const float* gsrc, unsigned lds_byte_off) {
    unsigned long long ga = (unsigned long long)(uintptr_t)gsrc;
    v4u g0;
    g0.x = 1u;                                                  // count=1, no gather
    g0.y = lds_byte_off;                                        // lds_addr
    g0.z = (unsigned)(ga & 0xFFFFFFFFu);                        // global_addr[31:0]
    g0.w = (unsigned)((ga >> 32) & 0x1FFFFFFu) | (2u << 30);    // global_addr[56:32] | type=2

    const unsigned td0 = TILE_ELEMS;   // tensor length dim0 (elements)
    const unsigned td1 = 1u;
    const unsigned tl0 = TILE_ELEMS;   // tile_dim0 (fits in 16 bits: 5120)
    v8i g1;
    g1[0] = (int)(2u << 16);                                        // data_size=2 -> 4 bytes
    g1[1] = (int)((td0 & 0xFFFFu) << 16);                           // tensor_dim0[15:0]
    g1[2] = (int)(((td0 >> 16) & 0xFFFFu) | ((td1 & 0xFFFFu) << 16));
    g1[3] = (int)(((td1 >> 16) & 0xFFFFu) | (tl0 << 16));           // tile_dim0
    g1[4] = 0;                                                      // tile_dim1=0, tile_dim2=0
    g1[5] = (int)td0;                                               // dim0 stride (unused, 1-D)
    g1[6] = 0;
    g1[7] = 0;
    v4i z4 = {0, 0, 0, 0};
    v8i z8 = {0, 0, 0, 0, 0, 0, 0, 0};
    __builtin_amdgcn_tensor_load_to_lds(g0, g1, z4, z4, z8, 0);
}

__global__ __launch_bounds__(THREADS)
void wce_main(const float* __restrict__ y_pred, const int* __restrict__ y_true,
              const float* __restrict__ adj,    const float* __restrict__ sw,
              float* __restrict__ partials) {
    __shared__ Smem sm;
    const int tid = threadIdx.x;
    const int bid = blockIdx.x;
    const long long base = (long long)bid * SAMPLES_PER_BLOCK;

    if (tid < NCLS * NCLS) sm.ratio[tid] = fmaxf(adj[tid], MIN_VALUE);

    // Prologue: wave 0 kicks the first tile DMA.
    if (tid < 32)
        tdm_load_tile(y_pred + base * NCLS, (unsigned)(uintptr_t)&sm.pred[0][0]);

    float numer = 0.0f, denom = 0.0f;
    int cur = 0;

    for (int t = 0; t < TILES_PER_BLOCK; ++t) {
        if (tid < 32) {
            __builtin_amdgcn_s_wait_tensorcnt(0);          // tile t landed in LDS
            if (t + 1 < TILES_PER_BLOCK)                   // prefetch t+1 into other buffer
                tdm_load_tile(y_pred + (base + (long long)(t + 1) * SAMPLES_PER_TILE) * NCLS,
                              (unsigned)(uintptr_t)&sm.pred[cur ^ 1][0]);
        }
        __syncthreads();                                   // tile t visible to all waves

        #pragma unroll
        for (int k = 0; k < SAMPLES_PER_THREAD; ++k) {
            const int sl = tid + k * THREADS;              // conflict-free: stride-5 floats
            const long long sg = base + (long long)t * SAMPLES_PER_TILE + sl;
            const float* x = &sm.pred[cur][sl * NCLS];
            const float x0 = x[0], x1 = x[1], x2 = x[2], x3 = x[3], x4 = x[4];

            // argmax, first-max wins (matches jnp.argmax)
            int pred = 0; float best = x0;
            if (x1 > best) { best = x1; pred = 1; }
            if (x2 > best) { best = x2; pred = 2; }
            if (x3 > best) { best = x3; pred = 3; }
            if (x4 > best) { best = x4; pred = 4; }

            const float s = __expf(x0 - best) + __expf(x1 - best) + __expf(x2 - best) +
                            __expf(x3 - best) + __expf(x4 - best);
            const float lse = __logf(s) + best;

            const int yt = y_true[sg];
            float xt = x0;
            xt = (yt == 1) ? x1 : xt;
            xt = (yt == 2) ? x2 : xt;
            xt = (yt == 3) ? x3 : xt;
            xt = (yt == 4) ? x4 : xt;

            const float w  = sw[sg];
            const float ce = lse - xt;
            numer += ce * sm.ratio[yt * NCLS + pred] * w;
            denom += w;
        }
        __syncthreads();                                   // everyone done with buf[cur]
        cur ^= 1;
    }

    // wave32 reduction, then cross-wave via LDS (deterministic)
    #pragma unroll
    for (int off = 16; off > 0; off >>= 1) {
        numer += __shfl_xor(numer, off, 32);
        denom += __shfl_xor(denom, off, 32);
    }
    const int wid = tid >> 5;
    if ((tid & 31) == 0) { sm.wred[wid] = numer; sm.wred[8 + wid] = denom; }
    __syncthreads();
    if (tid == 0) {
        float n = 0.0f, d = 0.0f;
        #pragma unroll
        for (int i = 0; i < THREADS / 32; ++i) { n += sm.wred[i]; d += sm.wred[8 + i]; }
        partials[bid]           = n;
        partials[NBLOCKS + bid] = d;
    }
}

// Stage 2: single wave reduces the 1024 block partials with the matrix unit:
// C += A(16x4 partials) x B(4x16 ones)  => accumulated row-sums, full f32.
__global__ __launch_bounds__(32)
void wce_reduce(const float* __restrict__ partials, float* __restrict__ out) {
    const int lane = threadIdx.x;
    v2f b; b.x = 1.0f; b.y = 1.0f;       // all-ones B (layout-agnostic)
    v8f cn = {}; v8f cd = {};

    #pragma unroll
    for (int i = 0; i < NBLOCKS / 64; ++i) {
        v2f an, ad;
        an.x = partials[i * 64 + lane];
        an.y = partials[i * 64 + 32 + lane];
        ad.x = partials[NBLOCKS + i * 64 + lane];
        ad.y = partials[NBLOCKS + i * 64 + 32 + lane];
        cn = __builtin_amdgcn_wmma_f32_16x16x4_f32(false, an, false, b, (short)0, cn, false, false);
        cd = __builtin_amdgcn_wmma_f32_16x16x4_f32(false, ad, false, b, (short)0, cd, false, false);
    }

    // every column of D holds rowsum[m]; lanes 0..15 carry M=0..7 in v[0..7],
    // lanes 16..31 carry M=8..15 -> per-lane sum + xor-16 shuffle = grand total
    float sn = cn[0] + cn[1] + cn[2] + cn[3] + cn[4] + cn[5] + cn[6] + cn[7];
    float sd = cd[0] + cd[1] + cd[2] + cd[3] + cd[4] + cd[5] + cd[6] + cd[7];
    sn += __shfl_xor(sn, 16, 32);
    sd += __shfl_xor(sd, 16, 32);
    if (lane == 0) out[0] = sn / sd;
}

extern "C" void kernel_launch(void* const* d_in, const int* in_sizes, int n_in,
                              void* d_out, int out_size, void* d_ws, size_t ws_size,
                              hipStream_t stream) {
    const float* y_pred = (const float*)d_in[0];
    const int*   y_true = (const int*)d_in[1];
    const float* adj    = (const float*)d_in[2];
    const float* sw     = (const float*)d_in[3];
    float* partials = (float*)d_ws;     // 2*NBLOCKS floats = 8 KB
    float* out      = (float*)d_out;

    wce_main<<<NBLOCKS, THREADS, 0, stream>>>(y_pred, y_true, adj, sw, partials);
    wce_reduce<<<1, 32, 0, stream>>>(partials, out);
}